// WiseNet_24678882083700
// MI455X (gfx1250) — compile-verified
//
#include <hip/hip_runtime.h>

// WiseNet seed-similarity kernel for gfx1250 (MI455X).
// sim[n,s] = clip(exp(-(||fA_n||^2 + ||fS_s||^2 - 2 fA_n.fS_s)/64), 1e-6, 1-1e-6)
// blobs[n] = argmax_s sim[n,s] + 1, zeroed when > bg_min_index.
// Core GEMM done with V_WMMA_F32_16X16X4_F32 (fp32 WMMA -> bit-faithful precision).

typedef float v2f __attribute__((ext_vector_type(2)));
typedef float v8f __attribute__((ext_vector_type(8)));

#define HH   512
#define WW   512
#define CC   64
#define SS   128
#define NPIX (HH * WW)

__global__ __launch_bounds__(256) void wisenet_sim_kernel(
    const float* __restrict__ O,        // [C, H*W]
    const int*   __restrict__ seed_y,   // [S]
    const int*   __restrict__ seed_x,   // [S]
    const int*   __restrict__ bg_p,     // [1]
    int*         __restrict__ blobs,    // [H*W]
    float*       __restrict__ sim)      // [H*W, S]
{
    // B operand tiles pre-swizzled into WMMA lane layout:
    // [tile t(8)][kgrp kk(16)][lane(32)][pair(2)] floats = 32 KB
    __shared__ float Bsw[8 * 16 * 32 * 2];
    __shared__ float bnorm[SS];

    const int tid  = threadIdx.x;
    const int lane = tid & 31;
    const int wave = tid >> 5;
    const int half = lane >> 4;     // 0: lanes 0-15, 1: lanes 16-31
    const int ml   = lane & 15;
    const int koff = half << 1;     // K sub-offset for A/B operand layout
    const int bg   = bg_p[0];

    // ---- Stage fS (seed embeddings) into LDS in WMMA-B layout ----
    for (int idx = tid; idx < SS * CC; idx += 256) {
        int s  = idx >> 6;          // seed index 0..127
        int ch = idx & 63;          // channel  0..63
        int pix = seed_y[s] * WW + seed_x[s];
        float v = O[(size_t)ch * NPIX + pix];
        int t  = s >> 4, sl = s & 15;
        int kk = ch >> 2, km = ch & 3;
        int ln = sl + ((km & 2) ? 16 : 0);   // lanes 0-15: K%4 in {0,1}; 16-31: {2,3}
        Bsw[(((t * 16 + kk) * 32 + ln) << 1) + (km & 1)] = v;
    }
    if (tid < SS) {                 // seed squared norms
        int pix = seed_y[tid] * WW + seed_x[tid];
        float acc = 0.0f;
        #pragma unroll
        for (int ch = 0; ch < CC; ++ch) {
            float v = O[(size_t)ch * NPIX + pix];
            acc += v * v;
        }
        bnorm[tid] = acc;
    }
    __syncthreads();

    const int n0 = blockIdx.x * 128 + wave * 16;   // this wave's 16-pixel strip

    // ---- Load A strip (16 pixels x 64 channels) in WMMA-A layout ----
    // 32-bit A 16x4: VGPR0 = K {0|2}, VGPR1 = K {1|3} across lane halves.
    v2f a[16];
    const float* Abase = O + n0 + ml;
    #pragma unroll
    for (int kk = 0; kk < 16; ++kk) {
        a[kk].x = Abase[(size_t)(4 * kk + koff)     * NPIX];
        a[kk].y = Abase[(size_t)(4 * kk + koff + 1) * NPIX];
    }

    // Pixel squared norms straight from A registers (no extra memory traffic).
    float part = 0.0f;
    #pragma unroll
    for (int kk = 0; kk < 16; ++kk) part += a[kk].x * a[kk].x + a[kk].y * a[kk].y;
    float anorm = part + __shfl_xor(part, 16, 32);  // lane m & m+16 both hold ||a_m||^2
    float an[8];
    #pragma unroll
    for (int r = 0; r < 8; ++r) an[r] = __shfl(anorm, r + half * 8, 32);

    float best[8];
    int   bidx[8];
    #pragma unroll
    for (int r = 0; r < 8; ++r) { best[r] = -1.0f; bidx[r] = 0; }

    // ---- 8 column tiles of 16 seeds; K=64 via 16x V_WMMA_F32_16X16X4_F32 ----
    #pragma unroll
    for (int t = 0; t < 8; ++t) {
        v8f c = {0.0f, 0.0f, 0.0f, 0.0f, 0.0f, 0.0f, 0.0f, 0.0f};
        #pragma unroll
        for (int kk = 0; kk < 16; ++kk) {
            v2f b = *(const v2f*)&Bsw[((t * 16 + kk) * 32 + lane) << 1]; // ds_load_b64
            c = __builtin_amdgcn_wmma_f32_16x16x4_f32(
                    /*neg_a=*/false, a[kk], /*neg_b=*/false, b,
                    /*c_mod=*/(short)0, c, /*reuse_a=*/false, /*reuse_b=*/false);
        }
        float bn = bnorm[t * 16 + ml];
        #pragma unroll
        for (int r = 0; r < 8; ++r) {
            // C/D layout: register r -> row M = r + 8*half, lane -> column ml
            float d2 = an[r] + bn - 2.0f * c[r];
            float sv = __expf(d2 * -0.015625f);            // exp(-d2/64)
            sv = fminf(fmaxf(sv, 1e-6f), 1.0f - 1e-6f);
            int n = n0 + r + half * 8;
            sim[(size_t)n * SS + t * 16 + ml] = sv;
            if (sv > best[r]) { best[r] = sv; bidx[r] = t * 16 + ml; } // strict >: first-index ties
        }
    }

    // ---- Per-row argmax butterfly within each 16-lane half ----
    #pragma unroll
    for (int r = 0; r < 8; ++r) {
        float bv = best[r];
        int   bi = bidx[r];
        #pragma unroll
        for (int off = 1; off <= 8; off <<= 1) {
            float ov = __shfl_xor(bv, off, 32);
            int   oi = __shfl_xor(bi, off, 32);
            if (ov > bv || (ov == bv && oi < bi)) { bv = ov; bi = oi; }
        }
        if (ml == r) {
            int blob = bi + 1;
            if (blob > bg) blob = 0;
            blobs[n0 + r + half * 8] = blob;
        }
    }
}

extern "C" void kernel_launch(void* const* d_in, const int* in_sizes, int n_in,
                              void* d_out, int out_size, void* d_ws, size_t ws_size,
                              hipStream_t stream) {
    (void)in_sizes; (void)n_in; (void)d_ws; (void)ws_size; (void)out_size;
    const float* O      = (const float*)d_in[0];
    const int*   seed_y = (const int*)d_in[1];
    const int*   seed_x = (const int*)d_in[2];
    const int*   bg     = (const int*)d_in[3];

    // Tuple output flattened in return order: blobs [H*W] then sim [H*W, S].
    int*   blobs = (int*)d_out;
    float* sim   = (float*)d_out + NPIX;

    const int blocks = NPIX / 128;   // 2048 blocks x 256 threads (8 wave32)
    wisenet_sim_kernel<<<blocks, 256, 0, stream>>>(O, seed_y, seed_x, bg, blobs, sim);
}